// TMCFAR_11759620456642
// MI455X (gfx1250) — compile-verified
//
#include <hip/hip_runtime.h>

#define H 512
#define W 512
#define TS 16
#define HR 4              // halo rows  (9//2)
#define HC 7              // halo cols  (15//2)
#define TR (TS + 2 * HR)  // 24
#define TC (TS + 2 * HC)  // 30

// ---- compare-exchange: x <- min, y <- max ---------------------------------
__device__ __forceinline__ void ce(float& x, float& y) {
    float mn = fminf(x, y);
    float mx = fmaxf(x, y);
    x = mn;
    y = mx;
}

// ---- Batcher odd-even mergesort, fully expanded at the C++ level ----------
// merge comparator chain: for (i = LO+R; i+R < LO+N; i += 2R) CE(i, i+R)
template <int I, int END, int R, int M>
__device__ __forceinline__ void oem_chain(float* v) {
    if constexpr (I + R < END) {
        ce(v[I], v[I + R]);
        oem_chain<I + M, END, R, M>(v);
    }
}

template <int LO, int N, int R>
__device__ __forceinline__ void oem_merge(float* v) {
    constexpr int M = R * 2;
    if constexpr (M < N) {
        oem_merge<LO, N, M>(v);      // even subsequence
        oem_merge<LO + R, N, M>(v);  // odd subsequence
        oem_chain<LO + R, LO + N, R, M>(v);
    } else {
        ce(v[LO], v[LO + R]);
    }
}

template <int LO, int N>
__device__ __forceinline__ void oem_sort(float* v) {
    if constexpr (N > 1) {
        constexpr int M = N / 2;
        oem_sort<LO, M>(v);
        oem_sort<LO + M, M>(v);
        oem_merge<LO, N, 1>(v);
    }
}

// ---- gather 126 reference cells (guard 3x3 excluded), constant indices ----
template <int P, int IDX>
__device__ __forceinline__ void gather(const float (*tile)[TC], int lr, int lc, float* v) {
    if constexpr (P < 135) {
        constexpr int dr = P / 15;
        constexpr int dc = P % 15;
        if constexpr (dr >= 3 && dr <= 5 && dc >= 6 && dc <= 8) {
            gather<P + 1, IDX>(tile, lr, lc, v);
        } else {
            v[IDX] = tile[lr + dr][lc + dc];
            gather<P + 1, IDX + 1>(tile, lr, lc, v);
        }
    }
}

// ---- sum of sorted ranks [31, 94) -----------------------------------------
template <int I>
__device__ __forceinline__ float tsum(const float* v) {
    if constexpr (I < 94) return v[I] + tsum<I + 1>(v);
    else return 0.0f;
}

__global__ __launch_bounds__(256, 2) void tmcfar_kernel(const float* __restrict__ in,
                                                        float* __restrict__ out) {
    __shared__ float tile[TR][TC];

    const int lr  = threadIdx.y;
    const int lc  = threadIdx.x;
    const int tid = lr * TS + lc;

    const int go_r = (int)blockIdx.y * TS - HR;
    const int go_c = (int)blockIdx.x * TS - HC;
    const float* chan = in + (size_t)blockIdx.z * (H * W);

    // Phase 1: zero-fill LDS tile (implements the zero padding at borders).
    for (int i = tid; i < TR * TC; i += 256)
        ((float*)tile)[i] = 0.0f;
    __syncthreads();

    // Phase 2: async global -> LDS copy of in-bounds tile elements (CDNA5 path).
    for (int i = tid; i < TR * TC; i += 256) {
        int rr = i / TC;
        int cc = i - rr * TC;
        int gr = go_r + rr;
        int gc = go_c + cc;
        if ((unsigned)gr < (unsigned)H && (unsigned)gc < (unsigned)W) {
            unsigned lds_off = (unsigned)(size_t)&tile[rr][cc];
            unsigned goff    = (unsigned)((gr * W + gc) * 4);
            asm volatile("global_load_async_to_lds_b32 %0, %1, %2"
                         :: "v"(lds_off), "v"(goff), "s"(chan)
                         : "memory");
        }
    }
    asm volatile("s_wait_asynccnt 0" ::: "memory");
    __syncthreads();

    // Phase 3: gather reference cells into registers + 2 pads to reach 128.
    float v[128];
    gather<0, 0>(tile, lr, lc, v);
    v[126] = __builtin_huge_valf();
    v[127] = __builtin_huge_valf();

    // Phase 4: Batcher odd-even mergesort (1471 compare-exchanges, in VGPRs).
    oem_sort<0, 128>(v);

    // Phase 5: trimmed mean over sorted ranks [31, 94) and divide.
    float s   = tsum<31>(v);
    float tm  = s / 63.0f;
    float val = tile[lr + HR][lc + HC];

    const int gy = (int)blockIdx.y * TS + lr;
    const int gx = (int)blockIdx.x * TS + lc;
    out[((size_t)blockIdx.z * H + gy) * W + gx] = val / tm;
}

extern "C" void kernel_launch(void* const* d_in, const int* in_sizes, int n_in,
                              void* d_out, int out_size, void* d_ws, size_t ws_size,
                              hipStream_t stream) {
    (void)in_sizes; (void)n_in; (void)d_ws; (void)ws_size; (void)out_size;
    const float* rd_map = (const float*)d_in[0];
    float* out = (float*)d_out;
    dim3 grid(W / TS, H / TS, 8);
    dim3 block(TS, TS, 1);
    tmcfar_kernel<<<grid, block, 0, stream>>>(rd_map, out);
}